// GraphAttentionLayer_46454366273963
// MI455X (gfx1250) — compile-verified
//
#include <hip/hip_runtime.h>
#include <hip/hip_bf16.h>
#include <stdint.h>

// ---- dims from the reference ----
#define BATCH 64
#define LL    1024
#define FIN   512
#define FOUT  512

// GEMM tiling
#define KC  32          // K-chunk staged in LDS
#define SA  36          // A LDS row stride (floats): 144B rows, b128-alignable, bank-conflict-free
#define SB  144         // B LDS row stride (floats): 2*144 % 64 == 32 -> conflict-free lane halves

typedef __attribute__((ext_vector_type(2))) float v2f;
typedef __attribute__((ext_vector_type(8))) float v8f;

__device__ __forceinline__ v8f wmma4(v2f a, v2f b, v8f c) {
    return __builtin_amdgcn_wmma_f32_16x16x4_f32(false, a, false, b, (short)0, c, false, false);
}

// ============================================================================
// FP32 WMMA GEMM: C[M,N] = A[M,K] * B[K,N], row-major, batched.
// Workgroup: 256 thr = 8 waves (2x4); each wave owns a 32x32 C block (2x2
// WMMA tiles) -> WG covers 64 rows x 128 cols.
// Both panels double-buffered in LDS and filled with CDNA5 async-to-LDS b128
// copies (ASYNCcnt); chunk c+1 staging overlaps chunk c's 32 WMMAs.
//   A panel: 64 x KC  (fragments -> single ds_load_b64 per tile row)
//   B panel: KC x 128 (fragments -> paired ds_load_2addr_b32)
// ============================================================================
__global__ __launch_bounds__(256)
void wmma_gemm_f32(const float* __restrict__ A, const float* __restrict__ B,
                   float* __restrict__ C, int K,
                   int lda, int ldb, int ldc,
                   long long strideA, long long strideB, long long strideC)
{
    __shared__ float As[2][64 * SA];   //  9216 B per buffer
    __shared__ float Bs[2][KC * SB];   // 18432 B per buffer

    const int batch = blockIdx.z;
    A += (long long)batch * strideA;
    B += (long long)batch * strideB;
    C += (long long)batch * strideC;

    const int tid  = threadIdx.x;
    const int wave = tid >> 5;
    const int lane = tid & 31;
    const int half = lane >> 4;      // 0: K={0,1}, 1: K={2,3}
    const int mn   = lane & 15;      // M for A-frags, N for B/C

    const int wr = wave >> 2;        // 0..1
    const int wc = wave & 3;         // 0..3

    const int rowBase = blockIdx.y * 64;
    const int colBase = blockIdx.x * 128;
    const int row0    = rowBase + wr * 32;
    const int col0    = colBase + wc * 32;

    v8f acc00 = {}, acc01 = {}, acc10 = {}, acc11 = {};

    // ---- async staging of one K-chunk into LDS buffer `buf` ----
    auto stage = [&](int buf, int k0) {
        // A: 64 rows x 32 floats = 512 float4 -> 2 shots/thread
#pragma unroll
        for (int i = 0; i < 2; ++i) {
            const int q    = tid + i * 256;
            const int row  = q >> 3;           // 8 float4 per 32-float row
            const int col4 = q & 7;
            const float*   gp = A + (long long)(rowBase + row) * lda + k0 + col4 * 4;
            const uint32_t lo = (uint32_t)(uintptr_t)&As[buf][row * SA + col4 * 4];
            asm volatile("global_load_async_to_lds_b128 %0, %1, off"
                         :: "v"(lo), "v"(gp) : "memory");
        }
        // B: 32 rows x 128 floats = 1024 float4 -> 4 shots/thread
#pragma unroll
        for (int i = 0; i < 4; ++i) {
            const int q    = tid + i * 256;
            const int row  = q >> 5;           // 32 float4 per 128-float row
            const int col4 = q & 31;
            const float*   gp = B + (long long)(k0 + row) * ldb + colBase + col4 * 4;
            const uint32_t lo = (uint32_t)(uintptr_t)&Bs[buf][row * SB + col4 * 4];
            asm volatile("global_load_async_to_lds_b128 %0, %1, off"
                         :: "v"(lo), "v"(gp) : "memory");
        }
    };

    const int nChunks = K / KC;
    stage(0, 0);

    for (int c = 0; c < nChunks; ++c) {
        asm volatile("s_wait_asynccnt 0x0" ::: "memory");  // chunk c landed (this wave)
        __syncthreads();                                   // ... and visible to all waves
        const int buf = c & 1;

        // overlap: kick off chunk c+1 into the other buffer
        if (c + 1 < nChunks) stage(buf ^ 1, (c + 1) * KC);

        const float* __restrict__ as = &As[buf][(wr * 32 + mn) * SA + 2 * half];
        const float* __restrict__ bs = &Bs[buf][(2 * half) * SB + wc * 32 + mn];

#pragma unroll
        for (int kk = 0; kk < KC; kk += 4) {
            v2f a0 = *(const v2f*)(as + kk);                 // rows row0+mn,    K pair
            v2f a1 = *(const v2f*)(as + kk + 16 * SA);       // rows row0+16+mn
            const float* bp = bs + kk * SB;
            v2f b0, b1;
            b0.x = bp[0];   b0.y = bp[SB];                   // cols +0..15
            b1.x = bp[16];  b1.y = bp[SB + 16];              // cols +16..31

            acc00 = wmma4(a0, b0, acc00);
            acc01 = wmma4(a0, b1, acc01);
            acc10 = wmma4(a1, b0, acc10);
            acc11 = wmma4(a1, b1, acc11);
        }
        __syncthreads();   // all waves done reading buf before it is re-staged
    }

    // ---- store: C VGPR v <-> row +v + 8*half, col = mn
    {
        float* __restrict__ Cr0 = C + (long long)(row0 +      8 * half) * ldc + col0 + mn;
        float* __restrict__ Cr1 = C + (long long)(row0 + 16 + 8 * half) * ldc + col0 + mn;
#pragma unroll
        for (int v = 0; v < 8; ++v) {
            Cr0[(long long)v * ldc]      = acc00[v];
            Cr0[(long long)v * ldc + 16] = acc01[v];
            Cr1[(long long)v * ldc]      = acc10[v];
            Cr1[(long long)v * ldc + 16] = acc11[v];
        }
    }
}

// ============================================================================
// e1[r] = dot(Wx[r,:], a[0:512]),  e2[r] = dot(Wx[r,:], a[512:1024])
// One wave32 per row, lane-strided loads + shfl_xor tree reduction.
// ============================================================================
__global__ __launch_bounds__(256)
void e12_kernel(const float* __restrict__ Wx, const float* __restrict__ a,
                float* __restrict__ e1, float* __restrict__ e2)
{
    const int row  = blockIdx.x * 8 + (threadIdx.x >> 5);   // [0, BATCH*LL)
    const int lane = threadIdx.x & 31;
    const float* __restrict__ w = Wx + (long long)row * FOUT;

    float s1 = 0.f, s2 = 0.f;
#pragma unroll
    for (int i = lane; i < FOUT; i += 32) {
        float v = w[i];
        s1 = fmaf(v, a[i], s1);
        s2 = fmaf(v, a[FOUT + i], s2);
    }
#pragma unroll
    for (int off = 16; off > 0; off >>= 1) {
        s1 += __shfl_xor(s1, off, 32);
        s2 += __shfl_xor(s2, off, 32);
    }
    if (lane == 0) { e1[row] = s1; e2[row] = s2; }
}

// ============================================================================
// attention[b,i,j] = softmax_j( leakyrelu(e1[b,i] + e2[b,j], 0.2) )
// One 256-thread block per (b,i) row; 4 elements per thread; two block
// reductions (max, sum) via wave shfl + LDS across the 8 waves.
// ============================================================================
__global__ __launch_bounds__(256)
void attn_softmax_kernel(const float* __restrict__ e1, const float* __restrict__ e2,
                         float* __restrict__ attn)
{
    const int row  = blockIdx.x;          // b*LL + i
    const int b    = row >> 10;
    const int lane = threadIdx.x & 31;
    const int wave = threadIdx.x >> 5;

    const float base = e1[row];
    const float* __restrict__ e2b = e2 + (long long)b * LL;

    __shared__ float sred[8];

    float vals[4];
    float m = -3.0e38f;
#pragma unroll
    for (int t = 0; t < 4; ++t) {
        const int j = threadIdx.x * 4 + t;
        float e = base + e2b[j];
        e = e > 0.f ? e : 0.2f * e;
        vals[t] = e;
        m = fmaxf(m, e);
    }
    // ---- block max ----
#pragma unroll
    for (int off = 16; off > 0; off >>= 1) m = fmaxf(m, __shfl_xor(m, off, 32));
    if (lane == 0) sred[wave] = m;
    __syncthreads();
    float bm = (lane < 8) ? sred[lane] : -3.0e38f;
#pragma unroll
    for (int off = 4; off > 0; off >>= 1) bm = fmaxf(bm, __shfl_xor(bm, off, 32));
    if (threadIdx.x == 0) sred[0] = bm;
    __syncthreads();
    const float rowmax = sred[0];
    __syncthreads();

    // ---- exp + block sum ----
    float s = 0.f;
#pragma unroll
    for (int t = 0; t < 4; ++t) {
        vals[t] = __expf(vals[t] - rowmax);
        s += vals[t];
    }
#pragma unroll
    for (int off = 16; off > 0; off >>= 1) s += __shfl_xor(s, off, 32);
    if (lane == 0) sred[wave] = s;
    __syncthreads();
    float bs = (lane < 8) ? sred[lane] : 0.f;
#pragma unroll
    for (int off = 4; off > 0; off >>= 1) bs += __shfl_xor(bs, off, 32);
    if (threadIdx.x == 0) sred[0] = bs;
    __syncthreads();
    const float inv = 1.0f / sred[0];

    float* __restrict__ out = attn + (long long)row * LL;
#pragma unroll
    for (int t = 0; t < 4; ++t)
        out[threadIdx.x * 4 + t] = vals[t] * inv;
}

// ============================================================================
// launcher
// ============================================================================
extern "C" void kernel_launch(void* const* d_in, const int* in_sizes, int n_in,
                              void* d_out, int out_size, void* d_ws, size_t ws_size,
                              hipStream_t stream)
{
    const float* x = (const float*)d_in[0];   // [B, L, FIN]
    const float* W = (const float*)d_in[1];   // [FIN, FOUT]
    const float* a = (const float*)d_in[2];   // [2*FOUT]

    float* h_prime = (float*)d_out;                                   // [B, L, FOUT]
    float* attn    = (float*)d_out + (long long)BATCH * LL * FOUT;    // [B, L, L]

    float* Wx = (float*)d_ws;                                         // [B, L, FOUT]
    float* e1 = Wx + (long long)BATCH * LL * FOUT;                    // [B*L]
    float* e2 = e1 + (long long)BATCH * LL;                           // [B*L]

    // 1) Wx = x @ W   (per batch: 1024x512 = 1024x512 @ 512x512; W shared)
    {
        dim3 grid(FOUT / 128, LL / 64, BATCH);
        wmma_gemm_f32<<<grid, 256, 0, stream>>>(
            x, W, Wx, /*K=*/FIN,
            /*lda=*/FIN, /*ldb=*/FOUT, /*ldc=*/FOUT,
            (long long)LL * FIN, 0LL, (long long)LL * FOUT);
    }

    // 2) e1, e2 = Wx @ a1, Wx @ a2
    e12_kernel<<<(BATCH * LL) / 8, 256, 0, stream>>>(Wx, a, e1, e2);

    // 3) attention = softmax(leakyrelu(e1[:,None] + e2[None,:]))
    attn_softmax_kernel<<<BATCH * LL, 256, 0, stream>>>(e1, e2, attn);

    // 4) h_prime = attention @ Wx   (per batch: 1024x512 = 1024x1024 @ 1024x512)
    {
        dim3 grid(FOUT / 128, LL / 64, BATCH);
        wmma_gemm_f32<<<grid, 256, 0, stream>>>(
            attn, Wx, h_prime, /*K=*/LL,
            /*lda=*/LL, /*ldb=*/FOUT, /*ldc=*/FOUT,
            (long long)LL * LL, (long long)LL * FOUT, (long long)LL * FOUT);
    }
}